// ExactCascade_21036749816535
// MI455X (gfx1250) — compile-verified
//
#include <hip/hip_runtime.h>
#include <hip/hip_bf16.h>

typedef float v2f __attribute__((ext_vector_type(2)));
typedef float v8f __attribute__((ext_vector_type(8)));

static __device__ __forceinline__ v8f wmma4(v2f a, v2f b, v8f c) {
  // D(16x16,f32) = A(16x4,f32) x B(4x16,f32) + C ; exact fp32 matrix op
  return __builtin_amdgcn_wmma_f32_16x16x4_f32(false, a, false, b, (short)0, c,
                                               false, false);
}

// ---------------------------------------------------------------------------
// Prep 1: W1eff[m] = T[m] (256x256) @ pw1[1:257,:] (256x512), stored directly
// in WMMA B-fragment layout:  W1s[(((m*32+ct)*64+ks)*32+lane)*2+slot]
//   where for element (n,c): ks=n>>2, lane=16*((n>>1)&1)+(c&15), slot=n&1.
// One wave per 16x16 output tile; 4096 tiles total.
// ---------------------------------------------------------------------------
__global__ __launch_bounds__(128) void prep_w1s_kernel(
    const float* __restrict__ T, const float* __restrict__ pw1,
    float* __restrict__ W1s) {
  int wid = (blockIdx.x << 2) + (threadIdx.x >> 5);
  int lane = threadIdx.x & 31;
  int hi = lane >> 4, lo = lane & 15;
  int m = wid >> 9, rem = wid & 511, nt = rem >> 5, ct = rem & 31;

  const float* Tm = T + ((size_t)(m * 256 + nt * 16 + lo)) * 256;
  const float* Pc = pw1 + 512 + ct * 16 + lo;  // skip row 0 (chi row)

  v8f acc = {};
#pragma unroll 4
  for (int ks = 0; ks < 64; ++ks) {
    int k0 = ks * 4 + hi * 2;
    v2f a = *(const v2f*)(Tm + k0);       // T[m][n][k0..k0+1]
    v2f b;
    b.x = Pc[k0 * 512];                   // pw1[1+k0][c]
    b.y = Pc[(k0 + 1) * 512];
    acc = wmma4(a, b, acc);
  }
#pragma unroll
  for (int v = 0; v < 8; ++v) {
    int n = nt * 16 + v + 8 * hi;         // C layout: row = v + 8*hi
    int ks2 = n >> 2, lane2 = 16 * ((n >> 1) & 1) + lo, slot = n & 1;
    W1s[(((size_t)(m * 32 + ct) * 64 + ks2) * 32 + lane2) * 2 + slot] = acc[v];
  }
}

// ---------------------------------------------------------------------------
// Prep 2: swizzle pw2 (512x256) into B-fragment layout
// ---------------------------------------------------------------------------
__global__ __launch_bounds__(256) void prep_pw2s_kernel(
    const float* __restrict__ pw2, float* __restrict__ pw2s) {
  int idx = blockIdx.x * 256 + threadIdx.x;  // 0..131071
  int k = idx >> 8, c = idx & 255;
  int ct = c >> 4, ks = k >> 2, lane = 16 * ((k >> 1) & 1) + (c & 15),
      slot = k & 1;
  pw2s[((ct * 128 + ks) * 32 + lane) * 2 + slot] = pw2[idx];
}

// ---------------------------------------------------------------------------
// Prep 3: pack the small 64x64 MLP weights (ew2, sw1, lw1, lw2) and the
// zero-padded 64x16 sw2 into B-fragment layout for the chi WMMA kernel.
// Wp layout: [ew2|sw1|lw1|lw2] @ 4096 floats each, then sw2pad @ 1024 floats.
// ---------------------------------------------------------------------------
__global__ __launch_bounds__(256) void prep_pack_kernel(
    const float* __restrict__ ew2, const float* __restrict__ sw1,
    const float* __restrict__ lw1, const float* __restrict__ lw2,
    const float* __restrict__ sw2, float* __restrict__ Wp) {
  int idx = blockIdx.x * 256 + threadIdx.x;  // 0..17407
  if (idx < 16384) {
    int which = idx >> 12, e = idx & 4095;
    int k = e >> 6, c = e & 63;  // weight is [k(64) x c(64)] row-major
    const float* src = (which == 0) ? ew2
                       : (which == 1) ? sw1
                       : (which == 2) ? lw1 : lw2;
    int ct = c >> 4, ks = k >> 2, lane = 16 * ((k >> 1) & 1) + (c & 15),
        slot = k & 1;
    Wp[which * 4096 + ((ct * 16 + ks) * 32 + lane) * 2 + slot] = src[k * 64 + c];
  } else {
    int e = idx - 16384;  // sw2 padded to 64x16
    int k = e >> 4, c = e & 15;
    int ks = k >> 2, lane = 16 * ((k >> 1) & 1) + c, slot = k & 1;
    Wp[4 * 4096 + (ks * 32 + lane) * 2 + slot] = (c < 8) ? sw2[k * 8 + c] : 0.f;
  }
}

// ---------------------------------------------------------------------------
// Chi phase (WMMA): z-chain + selector MLP, all on the matrix pipes.
// One wave owns one 16-row subtile; z/h tiles (16x64, A-frag layout) live in
// wave-private LDS slices -> zero barriers, 128 rows per 256-thread block.
// ---------------------------------------------------------------------------
static __device__ __forceinline__ void gemm64(const float* __restrict__ srcA,
                                              const v2f* __restrict__ Wf,
                                              const float* br, bool relu,
                                              float* __restrict__ dstA,
                                              int lane, int hi, int lo) {
#pragma unroll
  for (int t = 0; t < 4; ++t) {
    v8f acc = {};
#pragma unroll 4
    for (int ks = 0; ks < 16; ++ks)
      acc = wmma4(*(const v2f*)&srcA[(ks * 32 + lane) * 2],
                  Wf[(t * 16 + ks) * 32 + lane], acc);
    int k = t * 16 + lo;
    int base = ((k >> 2) * 32 + 16 * ((k >> 1) & 1)) * 2 + (k & 1);
#pragma unroll
    for (int v = 0; v < 8; ++v) {
      float val = acc[v] + br[t];
      if (relu) val = fmaxf(val, 0.f);
      dstA[base + (v + 8 * hi) * 2] = val;  // A-frag scatter
    }
  }
}

__global__ __launch_bounds__(256) void chi_wmma_kernel(
    const float* __restrict__ x, const float* __restrict__ ew1,
    const float* __restrict__ eb1, const float* __restrict__ eb2,
    const float* __restrict__ sb1, const float* __restrict__ sb2,
    const float* __restrict__ lb1, const float* __restrict__ lb2,
    const float* __restrict__ Wp, float* __restrict__ chi_all, int B) {
  __shared__ __align__(16) float zbuf[8][1024];  // 32 KB (A-frag, per wave)
  __shared__ __align__(16) float hbuf[8][1024];  // 32 KB (A-frag, per wave)
  int tid = threadIdx.x, w = tid >> 5, lane = tid & 31;
  int hi = lane >> 4, lo = lane & 15;
  int r0 = blockIdx.x * 128 + w * 16;
  float* zA = zbuf[w];
  float* hA = hbuf[w];
  const v2f* Wew2 = (const v2f*)(Wp);
  const v2f* Wsw1 = (const v2f*)(Wp + 4096);
  const v2f* Wlw1 = (const v2f*)(Wp + 8192);
  const v2f* Wlw2 = (const v2f*)(Wp + 12288);
  const v2f* Wsw2 = (const v2f*)(Wp + 16384);

  float eb2r[4], sb1r[4], lb1r[4], lb2r[4];
#pragma unroll
  for (int t = 0; t < 4; ++t) {
    int c = t * 16 + lo;
    eb2r[t] = eb2[c];
    sb1r[t] = sb1[c];
    lb1r[t] = lb1[c];
    lb2r[t] = lb2[c];
  }
  float sb2r = (lo < 8) ? sb2[lo] : 0.f;

  // h0 = relu(x * ew1 + eb1), built directly in A-frag layout
  for (int e = lane; e < 1024; e += 32) {
    int pair = e >> 1, slot = e & 1, ln = pair & 31, ks = pair >> 5;
    int row = ln & 15, k = ks * 4 + ((ln >> 4) << 1) + slot;
    hA[e] = fmaxf(fmaf(x[r0 + row], ew1[k], eb1[k]), 0.f);
  }
  // z0 = h0 @ ew2 + eb2
  gemm64(hA, Wew2, eb2r, false, zA, lane, hi, lo);

#pragma unroll 1
  for (int i = 0; i < 12; ++i) {
    // hz = relu(z @ sw1 + sb1)
    gemm64(zA, Wsw1, sb1r, true, hA, lane, hi, lo);
    // chi_i = hz @ sw2 + sb2  (valid cols 0..7)
    {
      v8f acc = {};
#pragma unroll 4
      for (int ks = 0; ks < 16; ++ks)
        acc = wmma4(*(const v2f*)&hA[(ks * 32 + lane) * 2],
                    Wsw2[ks * 32 + lane], acc);
      if (lo < 8) {
#pragma unroll
        for (int v = 0; v < 8; ++v)
          chi_all[(size_t)(i * B + r0 + v + 8 * hi) * 8 + lo] = acc[v] + sb2r;
      }
    }
    // h2 = relu(z @ lw1 + lb1) ; z = h2 @ lw2 + lb2
    gemm64(zA, Wlw1, lb1r, true, hA, lane, hi, lo);
    gemm64(hA, Wlw2, lb2r, false, zA, lane, hi, lo);
  }
}

// ---------------------------------------------------------------------------
// Phase B: persistent Phi evolution. One block = 32 batch rows (2 row-subtiles
// of 16), 8 waves. Phi tiles live in LDS (A-frag layout) across all 12 steps.
//   step: Hsum = sum_m relu(Phi @ W1eff[m] + chi*pw1[0] + pb1)  [32x512]
//         Phi  = Hsum @ pw2 + 8*pb2                             [32x256]
// Wave w owns Hsum cols [w*64, w*64+64) in GEMM1 (acc over m in regs) and Phi
// col-tiles {w, w+8} in GEMM2. Every B-fragment load from L2 feeds both
// row-subtiles (2 WMMAs) -> halves L2 traffic per FLOP.
// ---------------------------------------------------------------------------
__global__ __launch_bounds__(256) void phi_kernel(
    const float* __restrict__ W1s, const float* __restrict__ pw2s,
    const float* __restrict__ chi_all, const float* __restrict__ pw1,
    const float* __restrict__ pb1, const float* __restrict__ pb2,
    const float* __restrict__ u_mu, float* __restrict__ out, int B) {
  __shared__ __align__(16) float PhiA[2][64 * 32 * 2];    // 32 KB, A-frag
  __shared__ __align__(16) float HsumA[2][128 * 32 * 2];  // 64 KB, A-frag
  __shared__ float chi_t[32 * 8];

  int tid = threadIdx.x;
  int w = tid >> 5, lane = tid & 31;
  int hi = lane >> 4, lo = lane & 15;
  int r0 = blockIdx.x * 32;

  // Phi0 = broadcast u_mu (identical for every row -> row-independent fill)
  for (int p = tid; p < 2 * 2048; p += 256) {
    int s = p >> 11, pp = p & 2047;
    int ks = pp >> 5, ln = pp & 31;
    int k0 = ks * 4 + ((ln >> 4) << 1);
    PhiA[s][pp * 2] = u_mu[k0];
    PhiA[s][pp * 2 + 1] = u_mu[k0 + 1];
  }
  float pw10r[4], pb1r[4];
#pragma unroll
  for (int t = 0; t < 4; ++t) {
    int c = w * 64 + t * 16 + lo;
    pw10r[t] = pw1[c];  // pw1 row 0 (multiplies chi)
    pb1r[t] = pb1[c];
  }
  float pb2r[2];
#pragma unroll
  for (int j = 0; j < 2; ++j) pb2r[j] = 8.0f * pb2[(w + 8 * j) * 16 + lo];

  const v2f* W1v = (const v2f*)W1s;
  const v2f* P2v = (const v2f*)pw2s;

  for (int i = 0; i < 12; ++i) {
    chi_t[tid] = chi_all[(size_t)(i * B + r0 + (tid >> 3)) * 8 + (tid & 7)];
    __syncthreads();

    // ---- GEMM1: accumulate relu'd per-m results into hsum registers ----
    v8f hsum[2][4] = {};
#pragma unroll 1
    for (int m = 0; m < 8; ++m) {
      float chi_r[2][8];
#pragma unroll
      for (int s = 0; s < 2; ++s)
#pragma unroll
        for (int v = 0; v < 8; ++v)
          chi_r[s][v] = chi_t[(s * 16 + v + 8 * hi) * 8 + m];
      v8f acc[2][4] = {};
      const v2f* Wm = W1v + ((size_t)(m * 32 + w * 4) * 64) * 32 + lane;
#pragma unroll 2
      for (int ks = 0; ks < 64; ++ks) {
        v2f a0 = *(const v2f*)&PhiA[0][(ks * 32 + lane) * 2];
        v2f a1 = *(const v2f*)&PhiA[1][(ks * 32 + lane) * 2];
#pragma unroll
        for (int t = 0; t < 4; ++t) {
          v2f bf = Wm[(t * 64 + ks) * 32];
          acc[0][t] = wmma4(a0, bf, acc[0][t]);
          acc[1][t] = wmma4(a1, bf, acc[1][t]);
        }
      }
#pragma unroll
      for (int s = 0; s < 2; ++s)
#pragma unroll
        for (int t = 0; t < 4; ++t)
#pragma unroll
          for (int v = 0; v < 8; ++v)
            hsum[s][t][v] +=
                fmaxf(acc[s][t][v] + chi_r[s][v] * pw10r[t] + pb1r[t], 0.f);
    }
    // scatter hsum into LDS in A-frag layout for GEMM2
#pragma unroll
    for (int s = 0; s < 2; ++s)
#pragma unroll
      for (int t = 0; t < 4; ++t) {
        int k = w * 64 + t * 16 + lo;
        int ks2 = k >> 2, lb = 16 * ((k >> 1) & 1), slot = k & 1;
#pragma unroll
        for (int v = 0; v < 8; ++v) {
          int row = v + 8 * hi;
          HsumA[s][(ks2 * 32 + lb + row) * 2 + slot] = hsum[s][t][v];
        }
      }
    __syncthreads();

    // ---- GEMM2: Phi_new = Hsum @ pw2 + 8*pb2 ----
#pragma unroll
    for (int j = 0; j < 2; ++j) {
      int ct = w + 8 * j;
      v8f acc2[2] = {};
      const v2f* Pc = P2v + (size_t)ct * 128 * 32 + lane;
#pragma unroll 4
      for (int ks = 0; ks < 128; ++ks) {
        v2f bf = Pc[ks * 32];
        v2f a0 = *(const v2f*)&HsumA[0][(ks * 32 + lane) * 2];
        v2f a1 = *(const v2f*)&HsumA[1][(ks * 32 + lane) * 2];
        acc2[0] = wmma4(a0, bf, acc2[0]);
        acc2[1] = wmma4(a1, bf, acc2[1]);
      }
      if (i == 11) {
#pragma unroll
        for (int s = 0; s < 2; ++s)
#pragma unroll
          for (int v = 0; v < 8; ++v) {
            int row = v + 8 * hi, k = ct * 16 + lo;
            out[(size_t)(r0 + s * 16 + row) * 256 + k] = acc2[s][v] + pb2r[j];
          }
      } else {
#pragma unroll
        for (int s = 0; s < 2; ++s)
#pragma unroll
          for (int v = 0; v < 8; ++v) {
            int row = v + 8 * hi, k = ct * 16 + lo;
            int ks2 = k >> 2, lane2 = 16 * ((k >> 1) & 1) + row, slot = k & 1;
            PhiA[s][(ks2 * 32 + lane2) * 2 + slot] = acc2[s][v] + pb2r[j];
          }
      }
    }
    __syncthreads();
  }
}

// ---------------------------------------------------------------------------
extern "C" void kernel_launch(void* const* d_in, const int* in_sizes, int n_in,
                              void* d_out, int out_size, void* d_ws,
                              size_t ws_size, hipStream_t stream) {
  const float* x = (const float*)d_in[0];
  const float* ew1 = (const float*)d_in[1];
  const float* eb1 = (const float*)d_in[2];
  const float* ew2 = (const float*)d_in[3];
  const float* eb2 = (const float*)d_in[4];
  const float* lw1 = (const float*)d_in[5];
  const float* lb1 = (const float*)d_in[6];
  const float* lw2 = (const float*)d_in[7];
  const float* lb2 = (const float*)d_in[8];
  const float* sw1 = (const float*)d_in[9];
  const float* sb1 = (const float*)d_in[10];
  const float* sw2 = (const float*)d_in[11];
  const float* sb2 = (const float*)d_in[12];
  const float* pw1 = (const float*)d_in[25];
  const float* pb1 = (const float*)d_in[26];
  const float* pw2 = (const float*)d_in[27];
  const float* pb2 = (const float*)d_in[28];
  const float* T = (const float*)d_in[29];
  const float* u_mu = (const float*)d_in[30];

  const int B = in_sizes[0];  // 16384

  float* W1s = (float*)d_ws;                    // 1,048,576 floats (4 MB)
  float* pw2s = W1s + (1 << 20);                //   131,072 floats (512 KB)
  float* chi_all = pw2s + (1 << 17);            // 12*B*8 floats (6 MB)
  float* Wp = chi_all + (size_t)12 * B * 8;     //    17,408 floats (68 KB)

  prep_w1s_kernel<<<1024, 128, 0, stream>>>(T, pw1, W1s);
  prep_pw2s_kernel<<<512, 256, 0, stream>>>(pw2, pw2s);
  prep_pack_kernel<<<68, 256, 0, stream>>>(ew2, sw1, lw1, lw2, sw2, Wp);
  chi_wmma_kernel<<<B / 128, 256, 0, stream>>>(x, ew1, eb1, eb2, sb1, sb2, lb1,
                                               lb2, Wp, chi_all, B);
  phi_kernel<<<B / 32, 256, 0, stream>>>(W1s, pw2s, chi_all, pw1, pb1, pb2,
                                         u_mu, (float*)d_out, B);
}